// ModelTorch_32212254720223
// MI455X (gfx1250) — compile-verified
//
#include <hip/hip_runtime.h>
#include <stdint.h>

// Paged-KV extend-token slot assignment (reference: jnp scan + gather).
// Memory-bound: ~36 MB total traffic -> ~1.5us floor at 23.3 TB/s.
// No matrix structure => WMMA inapplicable; we exercise the CDNA5 async
// global->LDS data mover (ASYNCcnt) instead for the page-table staging.

#define PAGE_SHIFT 4
#define PAGE_MASK  15
#define SCAN_T     1024
// max new pages per seq: ceil(8207/16)+1 = 514; pad for safety
#define MAX_NEW_PAGES 544

// ---------------------------------------------------------------------------
// Kernel 1: per-batch exclusive scans of extend (token offsets) and new_pages
// (page offsets). Both packed into one u64 Hillis-Steele scan in LDS.
// ---------------------------------------------------------------------------
__global__ __launch_bounds__(SCAN_T)
void page_scan_kernel(const int* __restrict__ pre_lens,
                      const int* __restrict__ seq_lens,
                      int B,
                      int* __restrict__ out_start,
                      int* __restrict__ page_start) {
    __shared__ unsigned long long s[SCAN_T];
    const int tid = threadIdx.x;
    int carry_e = 0;   // running token total across chunks
    int carry_p = 0;   // running page total across chunks

    for (int base = 0; base < B; base += SCAN_T) {
        const int i = base + tid;
        int e = 0, np = 0;
        if (i < B) {
            const int p = pre_lens[i];
            const int q = seq_lens[i];
            e  = q - p;
            np = ((q + PAGE_MASK) >> PAGE_SHIFT) - ((p + PAGE_MASK) >> PAGE_SHIFT);
        }
        // pack: hi32 = pages, lo32 = tokens (both fit comfortably in 32 bits)
        s[tid] = (((unsigned long long)(unsigned)np) << 32) | (unsigned)e;
        __syncthreads();

        // inclusive Hillis-Steele scan over SCAN_T elements
        for (int d = 1; d < SCAN_T; d <<= 1) {
            unsigned long long add = (tid >= d) ? s[tid - d] : 0ULL;
            __syncthreads();
            s[tid] += add;
            __syncthreads();
        }

        const unsigned long long incl = s[tid];
        const unsigned long long tot  = s[SCAN_T - 1];
        if (i < B) {
            out_start[i]  = carry_e + (int)(unsigned)incl        - e;   // exclusive
            page_start[i] = carry_p + (int)(unsigned)(incl >> 32) - np; // exclusive
        }
        carry_e += (int)(unsigned)tot;
        carry_p += (int)(unsigned)(tot >> 32);
        __syncthreads();  // protect s[] before next chunk overwrites
    }
}

// ---------------------------------------------------------------------------
// Kernel 2: one block per sequence. Stage this sequence's slice of free_page
// into LDS via CDNA5 async global->LDS loads (ASYNCcnt), then emit token slot
// indices with coalesced b32 stores.
// ---------------------------------------------------------------------------
__global__ __launch_bounds__(256)
void fill_token_locs_kernel(const int* __restrict__ pre_lens,
                            const int* __restrict__ seq_lens,
                            const int* __restrict__ last_loc,
                            const int* __restrict__ free_page,
                            int fp_n,
                            const int* __restrict__ out_start,
                            const int* __restrict__ page_start,
                            int* __restrict__ out) {
    __shared__ int s_pages[MAX_NEW_PAGES];

    const int b        = blockIdx.x;
    const int pre      = pre_lens[b];
    const int seq      = seq_lens[b];
    const int ext      = seq - pre;
    const int npb      = (pre + PAGE_MASK) >> PAGE_SHIFT;   // pages already allocated
    const int boundary = npb << PAGE_SHIFT;                 // end of partial first page
    const int ll1      = last_loc[b] + 1;
    const int ostart   = out_start[b];
    const int pstart   = page_start[b];
    const int n_new    = ((seq + PAGE_MASK) >> PAGE_SHIFT) - npb;

    // --- stage free_page[pstart .. pstart+n_new) into LDS asynchronously ---
    for (int i = threadIdx.x; i < n_new; i += blockDim.x) {
        int gi = pstart + i;
        if (gi >= fp_n) gi = fp_n - 1;                      // matches jnp.clip safety
        unsigned int lds_addr = (unsigned int)(uintptr_t)(&s_pages[i]);
        const int*   gptr     = free_page + gi;
        // GLOBAL_LOAD_ASYNC_TO_LDS_B32: copies one dword from gptr to LDS at lds_addr; tracked on ASYNCcnt
        asm volatile("global_load_async_to_lds_b32 %0, %1, off"
                     :
                     : "v"(lds_addr), "v"(gptr)
                     : "memory");
    }
    asm volatile("s_wait_asynccnt 0x0" ::: "memory");       // this wave's async done
    __syncthreads();                                        // publish LDS to all waves

    // --- emit token slot indices; 16 consecutive lanes share one LDS entry ---
    for (int t = threadIdx.x; t < ext; t += blockDim.x) {
        const int pos = pre + t;
        int idx;
        if (pos < boundary) {
            idx = ll1 + t;                                  // continue partial page
        } else {
            const int pi = (pos >> PAGE_SHIFT) - npb;       // staged page index
            idx = (s_pages[pi] << PAGE_SHIFT) | (pos & PAGE_MASK);
        }
        out[ostart + t] = idx;                              // coalesced b32 store
    }
}

// ---------------------------------------------------------------------------
// Launch
// ---------------------------------------------------------------------------
extern "C" void kernel_launch(void* const* d_in, const int* in_sizes, int n_in,
                              void* d_out, int out_size, void* d_ws, size_t ws_size,
                              hipStream_t stream) {
    const int* pre_lens  = (const int*)d_in[0];
    const int* seq_lens  = (const int*)d_in[1];
    const int* last_loc  = (const int*)d_in[2];
    const int* free_page = (const int*)d_in[3];
    const int  B    = in_sizes[0];
    const int  fp_n = in_sizes[3];

    int* out_start  = (int*)d_ws;        // [B]
    int* page_start = out_start + B;     // [B]

    page_scan_kernel<<<1, SCAN_T, 0, stream>>>(pre_lens, seq_lens, B,
                                               out_start, page_start);

    // output is int32 in the reference; write raw int32 bits
    fill_token_locs_kernel<<<B, 256, 0, stream>>>(pre_lens, seq_lens, last_loc,
                                                  free_page, fp_n,
                                                  out_start, page_start,
                                                  (int*)d_out);
}